// trial_1932735283495
// MI455X (gfx1250) — compile-verified
//
#include <hip/hip_runtime.h>

// ---------------------------------------------------------------------------
// Reference reduces to: per (batch,channel) row of length T=1024,
//   s[t]   = -( |c[t]|*S + log1p(exp(-2*|c[t]|*S)) ) / S        (S = 1e5)
//   out[t] = max_{t' >= t} s[t']                                 (suffix max)
// Memory bound (512 KB total); one wave per batch row; TDM DMA -> padded LDS.
// corr term is bounded by ln2/1e5 = 6.9e-6, so hardware-TRANS exp/log
// (v_exp_f32 / v_log_f32) are exact-enough for it: error <= ~1e-11 absolute.
// ---------------------------------------------------------------------------

typedef unsigned int v4u __attribute__((ext_vector_type(4)));
typedef int          v8i __attribute__((ext_vector_type(8)));
typedef int          v4i __attribute__((ext_vector_type(4)));

#define T_LEN        1024
#define CHUNK        32                    // timesteps per lane
#define LANES        32
#define LDS_PAD_DW   2                     // TDM inserts 2 dwords per 256B chunk
#define LDS_STRIDE_F (CHUNK * 2 + LDS_PAD_DW)   // 66 floats per lane chunk

__device__ __forceinline__ float soft_and_neg_abs(float c) {
    // minish(stack([c,-c]), 1e5) replicated: m = max(c,-c)*scale (exact path),
    // corr = logsumexp residual (<= ln2), computed with hardware TRANS ops.
    float m    = fabsf(c) * 100000.0f;
    float corr = __logf(1.0f + __expf(-2.0f * m));   // v_exp_f32 / v_log_f32
    return -((m + corr) / 100000.0f);
}

__global__ __launch_bounds__(32)
void stl_suffix_kernel(const float* __restrict__ sig, float* __restrict__ out, int B)
{
    // 32 chunks * 264B + slack
    __shared__ float lds[LANES * LDS_STRIDE_F + 8];

    const int b    = blockIdx.x;
    const int lane = threadIdx.x & 31;

    // ---- Tensor Data Mover: DMA row (1024 x float2 = 8KB) global -> LDS ----
    // Low 32 bits of a flat shared pointer == wave-relative LDS byte offset.
    const unsigned long long gaddr =
        (unsigned long long)(const void*)(sig + (size_t)b * (2 * T_LEN));
    const unsigned lds_off = (unsigned)(unsigned long long)(const void*)lds;

#if __has_builtin(__builtin_amdgcn_tensor_load_to_lds)
    v4u g0;
    g0.x = 1u;                                     // count=1 (valid), not restore
    g0.y = lds_off;                                // lds_addr (bytes)
    g0.z = (unsigned)(gaddr & 0xffffffffull);      // global_addr[31:0]
    g0.w = (unsigned)((gaddr >> 32) & 0x01ffffffull) | 0x80000000u; // [56:32]|type=2

    v8i g1;
    g1[0] = (int)((2u << 16)        // data_size  = 4 bytes
                | (1u << 20)        // pad_enable = 1
                | (5u << 22)        // pad_interval: every 64 dwords (256B chunk)
                | (1u << 25));      // pad_amount  : 2 dwords (encoded N-1)
    g1[1] = (int)((unsigned)(2 * T_LEN) << 16);    // tensor_dim0 = 2048 (bits 63:48)
    g1[2] = (int)(1u << 16);                       // tensor_dim1 = 1    (bits 95:80)
    g1[3] = (int)((unsigned)(2 * T_LEN) << 16);    // tile_dim0   = 2048 (bits 127:112)
    g1[4] = 1;                                     // tile_dim1   = 1
    g1[5] = 2 * T_LEN;                             // tensor_dim0_stride low32
    g1[6] = 0;
    g1[7] = 0;

    v4i z4 = {0, 0, 0, 0};
#if defined(__clang_major__) && (__clang_major__ >= 23)
    v8i z8 = {0, 0, 0, 0, 0, 0, 0, 0};
    __builtin_amdgcn_tensor_load_to_lds(g0, g1, z4, z4, z8, 0);
#else
    __builtin_amdgcn_tensor_load_to_lds(g0, g1, z4, z4, 0);
#endif
    __builtin_amdgcn_s_wait_tensorcnt(0);
    __asm__ __volatile__("" ::: "memory");         // fence LDS reads behind the wait
#else
    // Fallback: cooperative coalesced copy with the same padded layout.
    const float* row = sig + (size_t)b * (2 * T_LEN);
    for (int i = lane; i < 2 * T_LEN; i += LANES) {
        lds[(i >> 6) * LDS_STRIDE_F + (i & 63)] = row[i];
    }
    __asm__ __volatile__("" ::: "memory");
#endif

    // ---- per-lane: load chunk (bank-conflict-free), soft-AND, suffix max ----
    const float* lchunk = lds + lane * LDS_STRIDE_F;   // 264B stride, 8B aligned

    float sx[CHUNK], sy[CHUNK];
#pragma unroll
    for (int k = 0; k < CHUNK; ++k) {
        float2 v = ((const float2*)lchunk)[k];
        sx[k] = soft_and_neg_abs(v.x);
        sy[k] = soft_and_neg_abs(v.y);
    }
#pragma unroll
    for (int k = CHUNK - 2; k >= 0; --k) {
        sx[k] = fmaxf(sx[k], sx[k + 1]);
        sy[k] = fmaxf(sy[k], sy[k + 1]);
    }

    // ---- cross-lane suffix max (wave32 Hillis-Steele via shfl_down) ----
    float mx = sx[0], my = sy[0];
#pragma unroll
    for (int d = 1; d < LANES; d <<= 1) {
        float ox = __shfl_down(mx, d, LANES);
        float oy = __shfl_down(my, d, LANES);
        if (lane + d < LANES) { mx = fmaxf(mx, ox); my = fmaxf(my, oy); }
    }
    float ex = __shfl_down(mx, 1, LANES);   // exclusive: max over later chunks
    float ey = __shfl_down(my, 1, LANES);
    if (lane == LANES - 1) { ex = -__builtin_inff(); ey = -__builtin_inff(); }

    // ---- combine + coalesced b128 stores ----
    float* outx = out + (size_t)b * T_LEN + lane * CHUNK;
    float* outy = out + (size_t)(B + b) * T_LEN + lane * CHUNK;
#pragma unroll
    for (int k = 0; k < CHUNK; k += 4) {
        float4 vx = make_float4(fmaxf(sx[k], ex), fmaxf(sx[k + 1], ex),
                                fmaxf(sx[k + 2], ex), fmaxf(sx[k + 3], ex));
        float4 vy = make_float4(fmaxf(sy[k], ey), fmaxf(sy[k + 1], ey),
                                fmaxf(sy[k + 2], ey), fmaxf(sy[k + 3], ey));
        *(float4*)(outx + k) = vx;
        *(float4*)(outy + k) = vy;
    }
}

extern "C" void kernel_launch(void* const* d_in, const int* in_sizes, int n_in,
                              void* d_out, int out_size, void* d_ws, size_t ws_size,
                              hipStream_t stream)
{
    const float* sig = (const float*)d_in[0];
    float*       out = (float*)d_out;
    const int B = in_sizes[0] / (2 * T_LEN);      // 32 for the reference shapes

    stl_suffix_kernel<<<dim3(B), dim3(32), 0, stream>>>(sig, out, B);

    (void)n_in; (void)out_size; (void)d_ws; (void)ws_size;
}